// RNNCell_59450937312098
// MI455X (gfx1250) — compile-verified
//
#include <hip/hip_runtime.h>
#include <hip/hip_bf16.h>
#include <cstddef>

typedef __attribute__((ext_vector_type(2))) float v2f;
typedef __attribute__((ext_vector_type(8))) float v8f;

#define B_ROWS 8192
#define T_LEN  2048
#define OBS    30

// ---------------------------------------------------------------------------
// Setup kernel: build block matrices P (32x32) and Q (32x32) from W_x, W_f.
// State padded to 32 (coeffs 0,1 are zero). One wave32, symbolic simulation:
//   C_0 = I; out_row = W'ᵀ C_i = row 31 of A^{i+1} = P[i,:]; C shifts up.
//   Q[i,j] = Wf * h[i-j], h[0]=1, h[d]=P[d-1,31].
// ---------------------------------------------------------------------------
__global__ void build_PQ_kernel(const float* __restrict__ Wx,
                                const float* __restrict__ Wf,
                                float* __restrict__ Pm,
                                float* __restrict__ Qm) {
    __shared__ float sP[32][32];
    const int k = threadIdx.x;            // 0..31, lane owns column k of C
    float wxp[32];
#pragma unroll
    for (int m = 0; m < 32; ++m) wxp[m] = (m < 2) ? 0.0f : Wx[m - 2];
    const float wf = Wf[0];

    float c[32];
#pragma unroll
    for (int m = 0; m < 32; ++m) c[m] = (m == k) ? 1.0f : 0.0f;

#pragma unroll
    for (int i = 0; i < 32; ++i) {
        float o = 0.0f;
#pragma unroll
        for (int m = 2; m < 32; ++m) o = fmaf(wxp[m], c[m], o);
        sP[i][k] = o;
        Pm[i * 32 + k] = o;
#pragma unroll
        for (int m = 0; m < 31; ++m) c[m] = c[m + 1];
        c[31] = o;
    }
    __syncthreads();   // single wave: ordering via DScnt, barrier is a NOP

#pragma unroll
    for (int i = 0; i < 32; ++i) {
        float q;
        if (k > i)       q = 0.0f;
        else if (k == i) q = wf;
        else             q = wf * sP[i - k - 1][31];
        Qm[i * 32 + k] = q;
    }
}

// ---------------------------------------------------------------------------
// Main kernel: one wave32 per 16-row tile. 64 sequential 32-step blocks:
//   OUT[16,32] = S[16,32] @ Pᵀ + U[16,32] @ Qᵀ   (fp32 WMMA 16x16x4, K-chunks)
//   S ← OUT (carried through wave-private LDS to convert D-layout → A-layout)
// ---------------------------------------------------------------------------
__global__ __launch_bounds__(32, 1) void ar_scan_kernel(
    const float* __restrict__ u,   // [B, T]
    const float* __restrict__ x0,  // [B, OBS]
    const float* __restrict__ Pm,  // [32,32]
    const float* __restrict__ Qm,  // [32,32]
    float* __restrict__ out)       // [B, T]
{
    __shared__ float st[16][32];   // state tile, [row][k]

    const int lane = threadIdx.x & 31;
    const int half = lane >> 4;    // 0: lanes 0-15, 1: lanes 16-31
    const int lm   = lane & 15;
    const int rowbase = blockIdx.x * 16;

    // -------- B operands for P and Q, resident in registers --------
    // B[k,n] tile j, chunk c: value = M[16j + n, 4c + k], lane holds n = lm,
    // k = 2*half + {0,1}  ->  v2f load of two consecutive row elements.
    v2f Pb[2][8], Qb[2][8];
#pragma unroll
    for (int j = 0; j < 2; ++j) {
#pragma unroll
        for (int c = 0; c < 8; ++c) {
            const int row = 16 * j + lm;
            const int col = 4 * c + 2 * half;
            Pb[j][c] = *(const v2f*)(Pm + row * 32 + col);
            Qb[j][c] = *(const v2f*)(Qm + row * 32 + col);
        }
    }

    // -------- initial state into LDS: s[row][k] = (k<2)?0:x0[row,k-2] ------
#pragma unroll
    for (int r = 0; r < 16; ++r) {
        float v = 0.0f;
        if (lane >= 2)
            v = x0[(size_t)(rowbase + r) * OBS + (lane - 2)];
        st[r][lane] = v;
    }

    // -------- software-pipelined u A-operand loads --------
    // A layout: lane holds (M=lm, K = 4c + 2*half + {0,1}) -> contiguous in t.
    const size_t urow = (size_t)(rowbase + lm) * T_LEN + (2 * half);
    v2f ua[8];
#pragma unroll
    for (int c = 0; c < 8; ++c)
        ua[c] = *(const v2f*)(u + urow + 4 * c);

    for (int tb = 0; tb < T_LEN; tb += 32) {
        // prefetch next block's u while this block computes
        v2f un[8];
        if (tb + 32 < T_LEN) {
#pragma unroll
            for (int c = 0; c < 8; ++c)
                un[c] = *(const v2f*)(u + urow + tb + 32 + 4 * c);
        }

        // state A-operand from LDS (ds_load_b64 per chunk)
        v2f sa[8];
#pragma unroll
        for (int c = 0; c < 8; ++c)
            sa[c] = *(const v2f*)(&st[lm][4 * c + 2 * half]);

        v8f acc0 = {};
        v8f acc1 = {};
#pragma unroll
        for (int c = 0; c < 8; ++c) {
            acc0 = __builtin_amdgcn_wmma_f32_16x16x4_f32(
                false, sa[c], false, Pb[0][c], (short)0, acc0, false, false);
            acc1 = __builtin_amdgcn_wmma_f32_16x16x4_f32(
                false, sa[c], false, Pb[1][c], (short)0, acc1, false, false);
        }
#pragma unroll
        for (int c = 0; c < 8; ++c) {
            acc0 = __builtin_amdgcn_wmma_f32_16x16x4_f32(
                false, ua[c], false, Qb[0][c], (short)0, acc0, false, false);
            acc1 = __builtin_amdgcn_wmma_f32_16x16x4_f32(
                false, ua[c], false, Qb[1][c], (short)0, acc1, false, false);
        }

        // D layout: vgpr r holds (M = r + 8*half, N = lm). Store to global
        // and write back as next block's state (LDS ops are in-order per wave).
#pragma unroll
        for (int r = 0; r < 8; ++r) {
            const int M = r + 8 * half;
            out[(size_t)(rowbase + M) * T_LEN + tb + lm]      = acc0[r];
            out[(size_t)(rowbase + M) * T_LEN + tb + 16 + lm] = acc1[r];
            st[M][lm]      = acc0[r];
            st[M][16 + lm] = acc1[r];
        }

#pragma unroll
        for (int c = 0; c < 8; ++c) ua[c] = un[c];
    }
}

// ---------------------------------------------------------------------------
extern "C" void kernel_launch(void* const* d_in, const int* in_sizes, int n_in,
                              void* d_out, int out_size, void* d_ws, size_t ws_size,
                              hipStream_t stream) {
    const float* u  = (const float*)d_in[0];   // [B, T, 1]
    const float* x0 = (const float*)d_in[1];   // [B, OBS]
    const float* Wx = (const float*)d_in[2];   // [OBS, 1]
    const float* Wf = (const float*)d_in[3];   // [1, 1]
    float* out = (float*)d_out;                // [B, T, 1]

    float* Pm = (float*)d_ws;        // 32*32 floats
    float* Qm = Pm + 32 * 32;        // 32*32 floats (8 KB total scratch)

    build_PQ_kernel<<<1, 32, 0, stream>>>(Wx, Wf, Pm, Qm);
    ar_scan_kernel<<<B_ROWS / 16, 32, 0, stream>>>(u, x0, Pm, Qm, out);
}